// LSTM_decoder_78486232367528
// MI455X (gfx1250) — compile-verified
//
#include <hip/hip_runtime.h>

typedef __attribute__((ext_vector_type(2))) float v2f;
typedef __attribute__((ext_vector_type(8))) float v8f;

constexpr int BB   = 64;    // batch
constexpr int TOUT = 25;    // output timesteps
constexpr int TENC = 50;    // encoder timesteps
constexpr int DD   = 1536;  // model dim (24*64)
constexpr int DD4  = 4 * DD;

__device__ __forceinline__ float sigmoidf_(float x) {
    return 1.0f / (1.0f + __expf(-x));
}

// ---------------------------------------------------------------------------
// Reductions: h0_init = mean_t(hidden), c0 = mean_t(cell) (two copies, each
// layer evolves its own in place), h1_init = mean of cat(global_t, hidden),
// n0 = mean_t(noise).
// ---------------------------------------------------------------------------
__global__ void __launch_bounds__(256)
means_kernel(const float* __restrict__ hs, const float* __restrict__ cs,
             const float* __restrict__ gts, const float* __restrict__ noise,
             float* __restrict__ h0_init, float* __restrict__ c0a,
             float* __restrict__ c0b, float* __restrict__ h1_init,
             float* __restrict__ n0)
{
    int idx = blockIdx.x * blockDim.x + threadIdx.x;
    if (idx >= BB * DD) return;
    int b = idx / DD, d = idx % DD;
    const float* ph = hs    + (size_t)b * TENC * DD + d;
    const float* pc = cs    + (size_t)b * TENC * DD + d;
    const float* pn = noise + (size_t)b * TENC * DD + d;
    float sh = 0.f, sc = 0.f, sn = 0.f;
    for (int t = 0; t < TENC; ++t) {
        sh += ph[t * DD];
        sc += pc[t * DD];
        sn += pn[t * DD];
    }
    h0_init[idx] = sh * (1.0f / TENC);
    float c0 = sc * (1.0f / TENC);
    c0a[idx] = c0;
    c0b[idx] = c0;
    h1_init[idx] = (sh + gts[idx]) * (1.0f / (TENC + 1));
    n0[idx] = sn * (1.0f / TENC);
}

// ---------------------------------------------------------------------------
// Core fp32 WMMA 4-gate tile (16 rows of A): acc[g] += A * W[g*DD+n0 ..]^T.
// A fragment (16x4 f32): lane<16 holds K=k,k+1 ; lane>=16 holds K=k+2,k+3.
// B fragment (4x16 f32): lane = output column; same K split across halves.
// Both operands are contiguous v2f loads (K-major rows).
// ---------------------------------------------------------------------------
__device__ __forceinline__ void gemm4_tile(const float* __restrict__ A, int astride,
                                           const float* __restrict__ W, int n0,
                                           int lane, v8f acc[4])
{
    const int lr = lane & 15;
    const int hi = lane >> 4;
    const float* arow = A + (size_t)lr * astride + 2 * hi;
    const float* w0 = W + (size_t)(0 * DD + n0 + lr) * DD + 2 * hi;
    const float* w1 = W + (size_t)(1 * DD + n0 + lr) * DD + 2 * hi;
    const float* w2 = W + (size_t)(2 * DD + n0 + lr) * DD + 2 * hi;
    const float* w3 = W + (size_t)(3 * DD + n0 + lr) * DD + 2 * hi;
#pragma unroll 8
    for (int k = 0; k < DD; k += 4) {
        v2f a  = *(const v2f*)(arow + k);
        v2f b0 = *(const v2f*)(w0 + k);
        v2f b1 = *(const v2f*)(w1 + k);
        v2f b2 = *(const v2f*)(w2 + k);
        v2f b3 = *(const v2f*)(w3 + k);
        acc[0] = __builtin_amdgcn_wmma_f32_16x16x4_f32(false, a, false, b0, (short)0, acc[0], false, false);
        acc[1] = __builtin_amdgcn_wmma_f32_16x16x4_f32(false, a, false, b1, (short)0, acc[1], false, false);
        acc[2] = __builtin_amdgcn_wmma_f32_16x16x4_f32(false, a, false, b2, (short)0, acc[2], false, false);
        acc[3] = __builtin_amdgcn_wmma_f32_16x16x4_f32(false, a, false, b3, (short)0, acc[3], false, false);
    }
}

// ---------------------------------------------------------------------------
// Time-parallel pre-gates: pre[m, g*DD+n] = X[m,:]·W[g*DD+n,:] + b_ih + b_hh.
// X is [M, DD] row-major (M = BB*TOUT = 1600). M is abundant here, so each
// wave computes TWO 16-row A fragments against the SAME four B (weight)
// fragments: 6 loads : 8 WMMAs per k-step (vs 5:4), halving weight traffic.
// Block = 8 waves along n (128 cols per gate). Grid = (DD/128, M/32).
// ---------------------------------------------------------------------------
__global__ void __launch_bounds__(256)
pregates_kernel(const float* __restrict__ X, const float* __restrict__ W,
                const float* __restrict__ b_ih, const float* __restrict__ b_hh,
                float* __restrict__ pre)
{
    int lane = threadIdx.x & 31, wave = threadIdx.x >> 5;
    int n0 = blockIdx.x * 128 + wave * 16;
    int m0 = blockIdx.y * 32;
    const int lr = lane & 15;
    const int hi = lane >> 4;

    const float* a0row = X + (size_t)(m0 + lr) * DD + 2 * hi;
    const float* a1row = a0row + (size_t)16 * DD;
    const float* w0 = W + (size_t)(0 * DD + n0 + lr) * DD + 2 * hi;
    const float* w1 = W + (size_t)(1 * DD + n0 + lr) * DD + 2 * hi;
    const float* w2 = W + (size_t)(2 * DD + n0 + lr) * DD + 2 * hi;
    const float* w3 = W + (size_t)(3 * DD + n0 + lr) * DD + 2 * hi;

    v8f acc0[4] = {};
    v8f acc1[4] = {};
#pragma unroll 4
    for (int k = 0; k < DD; k += 4) {
        v2f a0 = *(const v2f*)(a0row + k);
        v2f a1 = *(const v2f*)(a1row + k);
        v2f b0 = *(const v2f*)(w0 + k);
        v2f b1 = *(const v2f*)(w1 + k);
        v2f b2 = *(const v2f*)(w2 + k);
        v2f b3 = *(const v2f*)(w3 + k);
        acc0[0] = __builtin_amdgcn_wmma_f32_16x16x4_f32(false, a0, false, b0, (short)0, acc0[0], false, false);
        acc1[0] = __builtin_amdgcn_wmma_f32_16x16x4_f32(false, a1, false, b0, (short)0, acc1[0], false, false);
        acc0[1] = __builtin_amdgcn_wmma_f32_16x16x4_f32(false, a0, false, b1, (short)0, acc0[1], false, false);
        acc1[1] = __builtin_amdgcn_wmma_f32_16x16x4_f32(false, a1, false, b1, (short)0, acc1[1], false, false);
        acc0[2] = __builtin_amdgcn_wmma_f32_16x16x4_f32(false, a0, false, b2, (short)0, acc0[2], false, false);
        acc1[2] = __builtin_amdgcn_wmma_f32_16x16x4_f32(false, a1, false, b2, (short)0, acc1[2], false, false);
        acc0[3] = __builtin_amdgcn_wmma_f32_16x16x4_f32(false, a0, false, b3, (short)0, acc0[3], false, false);
        acc1[3] = __builtin_amdgcn_wmma_f32_16x16x4_f32(false, a1, false, b3, (short)0, acc1[3], false, false);
    }

    int n = n0 + lr;
    float bias[4];
#pragma unroll
    for (int g = 0; g < 4; ++g) bias[g] = b_ih[g * DD + n] + b_hh[g * DD + n];
#pragma unroll
    for (int r = 0; r < 8; ++r) {
        int m = m0 + r + hi * 8;
        float* prow0 = pre + (size_t)m * DD4;
        float* prow1 = pre + (size_t)(m + 16) * DD4;
#pragma unroll
        for (int g = 0; g < 4; ++g) {
            prow0[g * DD + n] = acc0[g][r] + bias[g];
            prow1[g * DD + n] = acc1[g][r] + bias[g];
        }
    }
}

// ---------------------------------------------------------------------------
// One recurrent LSTM step: gates = pre_t + h_prev @ W_hh^T, then fused
// i/f/o = sigmoid, g = tanh, c = f*c + i*g (in place), h = o*tanh(c).
// Each wave owns all four gate tiles for its (m,n) block, so the nonlinearity
// is applied in-register with no extra pass. M=64 and the time chain is
// serial, so we keep the 16-row tile for maximum wave parallelism.
// Grid = (DD/128, BB/16).
// ---------------------------------------------------------------------------
__global__ void __launch_bounds__(256)
lstm_step_kernel(const float* __restrict__ h_prev, int hstride,
                 const float* __restrict__ pre_t, int prestride,
                 const float* __restrict__ W_hh,
                 float* __restrict__ c_state,
                 float* __restrict__ h_out, int hostride,
                 float* __restrict__ c_out, int costride)
{
    int lane = threadIdx.x & 31, wave = threadIdx.x >> 5;
    int n0 = blockIdx.x * 128 + wave * 16;
    int m0 = blockIdx.y * 16;
    v8f acc[4] = {};
    gemm4_tile(h_prev + (size_t)m0 * hstride, hstride, W_hh, n0, lane, acc);
    int lr = lane & 15, hi = lane >> 4;
    int n = n0 + lr;
#pragma unroll
    for (int r = 0; r < 8; ++r) {
        int m = m0 + r + hi * 8;                       // batch row
        const float* prow = pre_t + (size_t)m * prestride;
        float gi = acc[0][r] + prow[0 * DD + n];
        float gf = acc[1][r] + prow[1 * DD + n];
        float gg = acc[2][r] + prow[2 * DD + n];
        float go = acc[3][r] + prow[3 * DD + n];
        float i = sigmoidf_(gi);
        float f = sigmoidf_(gf);
        float o = sigmoidf_(go);
        float g = tanhf(gg);
        float* cp = c_state + (size_t)m * DD + n;
        float cn = f * (*cp) + i * g;
        *cp = cn;
        h_out[(size_t)m * hostride + n] = o * tanhf(cn);
        if (c_out) c_out[(size_t)m * costride + n] = cn;
    }
}

// ---------------------------------------------------------------------------
// Noise chain step: Nout[64,1536] = Nin[64,1536] @ Noise[1536,1536].
// Noise is K-major over rows -> B fragment needs two strided dword loads.
// ---------------------------------------------------------------------------
__global__ void __launch_bounds__(256)
noise_gemm_kernel(const float* __restrict__ Nin, const float* __restrict__ NoiseM,
                  float* __restrict__ Nout)
{
    int lane = threadIdx.x & 31, wave = threadIdx.x >> 5;
    int n0 = blockIdx.x * 128 + wave * 16;
    int m0 = blockIdx.y * 16;
    int lr = lane & 15, hi = lane >> 4;
    const float* arow = Nin + (size_t)(m0 + lr) * DD + 2 * hi;
    const float* bcol = NoiseM + (size_t)(2 * hi) * DD + n0 + lr;
    v8f acc = {};
#pragma unroll 4
    for (int k = 0; k < DD; k += 4) {
        v2f a = *(const v2f*)(arow + k);
        v2f b;
        b.x = bcol[(size_t)(k + 0) * DD];
        b.y = bcol[(size_t)(k + 1) * DD];
        acc = __builtin_amdgcn_wmma_f32_16x16x4_f32(false, a, false, b, (short)0, acc, false, false);
    }
    int n = n0 + lr;
#pragma unroll
    for (int r = 0; r < 8; ++r) {
        int m = m0 + r + hi * 8;
        Nout[(size_t)m * DD + n] = acc[r];
    }
}

// out[b,t,d] = 0.8*h1_seq[b,t,d] + 0.2*nseq[t,b,d]
__global__ void __launch_bounds__(256)
blend_kernel(const float* __restrict__ h1_seq, const float* __restrict__ nseq,
             float* __restrict__ out)
{
    int idx = blockIdx.x * blockDim.x + threadIdx.x;
    if (idx >= BB * TOUT * DD) return;
    int d = idx % DD;
    int bt = idx / DD;
    int t = bt % TOUT;
    int b = bt / TOUT;
    out[idx] = 0.8f * h1_seq[idx] + 0.2f * nseq[((size_t)t * BB + b) * DD + d];
}

// ---------------------------------------------------------------------------
extern "C" void kernel_launch(void* const* d_in, const int* in_sizes, int n_in,
                              void* d_out, int out_size, void* d_ws, size_t ws_size,
                              hipStream_t stream)
{
    (void)in_sizes; (void)n_in; (void)out_size; (void)ws_size;
    const float* hs    = (const float*)d_in[0];
    const float* cs    = (const float*)d_in[1];
    const float* gts   = (const float*)d_in[2];
    const float* p     = (const float*)d_in[3];   // [BB, TOUT, DD] == A of layer-0 GEMM
    const float* noise = (const float*)d_in[4];
    const float* W_ih0 = (const float*)d_in[5];
    const float* W_hh0 = (const float*)d_in[6];
    const float* b_ih0 = (const float*)d_in[7];
    const float* b_hh0 = (const float*)d_in[8];
    const float* W_ih1 = (const float*)d_in[9];
    const float* W_hh1 = (const float*)d_in[10];
    const float* b_ih1 = (const float*)d_in[11];
    const float* b_hh1 = (const float*)d_in[12];
    const float* NoiseM = (const float*)d_in[13];

    float* ws = (float*)d_ws;
    float* h0_init = ws; ws += BB * DD;
    float* c0a     = ws; ws += BB * DD;
    float* c0b     = ws; ws += BB * DD;
    float* h1_init = ws; ws += BB * DD;
    float* n0      = ws; ws += BB * DD;
    float* pre0    = ws; ws += (size_t)BB * TOUT * DD4;   // [B*T, 4D]
    float* pre1    = ws; ws += (size_t)BB * TOUT * DD4;
    float* h0_seq  = ws; ws += (size_t)BB * TOUT * DD;    // [B, T, D]
    float* h1_seq  = ws; ws += (size_t)BB * TOUT * DD;
    float* nseq    = ws; ws += (size_t)BB * TOUT * DD;    // [T, B, D]

    float* out_pre = (float*)d_out;
    float* out_c   = out_pre + (size_t)BB * TOUT * DD;

    dim3 blk(256);

    means_kernel<<<(BB * DD + 255) / 256, blk, 0, stream>>>(
        hs, cs, gts, noise, h0_init, c0a, c0b, h1_init, n0);

    // Layer-0 input pre-gates: one big M=1600 GEMM (time-parallel).
    pregates_kernel<<<dim3(DD / 128, (BB * TOUT) / 32), blk, 0, stream>>>(
        p, W_ih0, b_ih0, b_hh0, pre0);

    for (int t = 0; t < TOUT; ++t) {
        const float* hp = (t == 0) ? h0_init : h0_seq + (size_t)(t - 1) * DD;
        int hstride = (t == 0) ? DD : TOUT * DD;
        lstm_step_kernel<<<dim3(DD / 128, BB / 16), blk, 0, stream>>>(
            hp, hstride, pre0 + (size_t)t * DD4, TOUT * DD4, W_hh0, c0a,
            h0_seq + (size_t)t * DD, TOUT * DD, nullptr, 0);
    }

    // Layer-1 input pre-gates from full h0_seq: second big M=1600 GEMM.
    pregates_kernel<<<dim3(DD / 128, (BB * TOUT) / 32), blk, 0, stream>>>(
        h0_seq, W_ih1, b_ih1, b_hh1, pre1);

    for (int t = 0; t < TOUT; ++t) {
        const float* hp = (t == 0) ? h1_init : h1_seq + (size_t)(t - 1) * DD;
        int hstride = (t == 0) ? DD : TOUT * DD;
        lstm_step_kernel<<<dim3(DD / 128, BB / 16), blk, 0, stream>>>(
            hp, hstride, pre1 + (size_t)t * DD4, TOUT * DD4, W_hh1, c0b,
            h1_seq + (size_t)t * DD, TOUT * DD,
            out_c + (size_t)t * DD, TOUT * DD);
    }

    // Noise chain (serial 25x small GEMM).
    for (int t = 0; t < TOUT; ++t) {
        const float* np = (t == 0) ? n0 : nseq + (size_t)(t - 1) * BB * DD;
        noise_gemm_kernel<<<dim3(DD / 128, BB / 16), blk, 0, stream>>>(
            np, NoiseM, nseq + (size_t)t * BB * DD);
    }

    blend_kernel<<<(BB * TOUT * DD + 255) / 256, blk, 0, stream>>>(
        h1_seq, nseq, out_pre);
}